// DynamicGraphAttention_14242111553594
// MI455X (gfx1250) — compile-verified
//
#include <hip/hip_runtime.h>

typedef __attribute__((ext_vector_type(2))) float v2f;
typedef __attribute__((ext_vector_type(8))) float v8f;

#define N_NODES 64
#define T_DIM   4096
#define HIDDEN  16
#define NEG_INF -1000000000.0f
#define ATT_SCALE 0.25f   // HIDDEN^-0.5 = 16^-0.5
#define W_PITCH 65        // 64 + 1 pad: avoid 64-bank-period LDS conflicts

// One block per batch element. 256 threads = 8 wave32 waves.
__launch_bounds__(256, 2)
__global__ void dga_kernel(const float* __restrict__ x,
                           const float* __restrict__ wq,
                           const float* __restrict__ bq,
                           const float* __restrict__ wk,
                           const float* __restrict__ bk,
                           float* __restrict__ out1,   // (256,1,64,4096)
                           float* __restrict__ out2)   // (256,64,64)
{
    __shared__ float s_e[N_NODES];              // per-row variance
    __shared__ float s_w[N_NODES * W_PITCH];    // attn weights, padded pitch

    const int b    = blockIdx.x;
    const int tid  = threadIdx.x;
    const int lane = tid & 31;
    const int wave = tid >> 5;

    const float* xs = x + (size_t)b * N_NODES * T_DIM;

    // ---------------- Phase A: row variances (ddof=1) ----------------
    // Each wave reduces 8 rows; each lane strides float4 over a row.
    for (int rr = 0; rr < 8; ++rr) {
        const int r = wave * 8 + rr;
        const float4* p = (const float4*)(xs + r * T_DIM);
        float sum = 0.f, sq = 0.f;
        for (int k = lane; k < T_DIM / 4; k += 32) {
            float4 v = p[k];
            sum += v.x + v.y + v.z + v.w;
            sq  += v.x * v.x + v.y * v.y + v.z * v.z + v.w * v.w;
        }
#pragma unroll
        for (int off = 16; off >= 1; off >>= 1) {
            sum += __shfl_xor(sum, off, 32);
            sq  += __shfl_xor(sq,  off, 32);
        }
        if (lane == 0) {
            // var = (sumsq - sum^2/N) / (N-1)
            s_e[r] = (sq - sum * sum * (1.0f / T_DIM)) * (1.0f / (T_DIM - 1));
        }
    }
    __syncthreads();

    // ---------------- Phase B: closed-form attn + softmax ----------------
    // attn[i,j] = SCALE*(dww*e_i*e_j + dwb*e_i + dbw*e_j + dbb), diag = -1e9
    float dww = 0.f, dwb = 0.f, dbw = 0.f, dbb = 0.f;
#pragma unroll
    for (int h = 0; h < HIDDEN; ++h) {
        const float a = wq[h], c = bq[h], d = wk[h], f = bk[h];
        dww += a * d; dwb += a * f; dbw += c * d; dbb += c * f;
    }
    if (tid < N_NODES) {
        const int i = tid;
        const float ei = s_e[i];
        float m = NEG_INF;
        for (int j = 0; j < N_NODES; ++j) {
            const float ej = s_e[j];
            const float a = (j == i) ? NEG_INF
                : ATT_SCALE * (dww * ei * ej + dwb * ei + dbw * ej + dbb);
            m = fmaxf(m, a);
        }
        float denom = 0.f;
        for (int j = 0; j < N_NODES; ++j) {
            const float ej = s_e[j];
            const float a = (j == i) ? NEG_INF
                : ATT_SCALE * (dww * ei * ej + dwb * ei + dbw * ej + dbb);
            denom += __expf(a - m);
        }
        const float inv = 1.0f / denom;
        float* o2 = out2 + (size_t)b * N_NODES * N_NODES + (size_t)i * N_NODES;
        for (int j = 0; j < N_NODES; ++j) {
            const float ej = s_e[j];
            const float a = (j == i) ? NEG_INF
                : ATT_SCALE * (dww * ei * ej + dwb * ei + dbw * ej + dbb);
            const float w = __expf(a - m) * inv;
            s_w[i * W_PITCH + j] = w;
            o2[j] = w;
        }
    }
    __syncthreads();

    // ---------------- Phase C: out = W @ xs + x via v_wmma_f32_16x16x4_f32 ----
    // A (16x4 f32): lanes 0-15 hold M=lane, K={4k,4k+1}; lanes 16-31 K={4k+2,4k+3}
    // B (4x16 f32): lanes 0-15 hold N=lane, K={4k,4k+1}; lanes 16-31 K={4k+2,4k+3}
    // C/D (16x16 f32): vgpr v: lanes 0-15 -> M=v, lanes 16-31 -> M=v+8; N=lane&15
    const int  halflo = (lane < 16);
    const int  joff   = halflo ? 0 : 2;
    const int  n15    = lane & 15;
    const int  mbase  = halflo ? 0 : 8;
    const size_t obase = (size_t)b * N_NODES * T_DIM;

    for (int tt = wave; tt < T_DIM / 16; tt += 8) {
        const int t0 = tt * 16 + n15;

        // Hoist B fragments for this 64x16 xs tile ONCE (read xs once, not 4x).
        v2f bfrag[16];
#pragma unroll
        for (int k = 0; k < 16; ++k) {
            const int jrow = 4 * k + joff;
            bfrag[k].x = xs[jrow * T_DIM + t0];
            bfrag[k].y = xs[(jrow + 1) * T_DIM + t0];
        }

        for (int it = 0; it < 4; ++it) {
            // A fragments for this 16-row tile of W from LDS (padded -> no conflicts).
            const int arow = it * 16 + n15;
            v2f afrag[16];
#pragma unroll
            for (int k = 0; k < 16; ++k) {
                afrag[k].x = s_w[arow * W_PITCH + 4 * k + joff];
                afrag[k].y = s_w[arow * W_PITCH + 4 * k + joff + 1];
            }
            v8f acc = {};
#pragma unroll
            for (int k = 0; k < 16; ++k) {
                acc = __builtin_amdgcn_wmma_f32_16x16x4_f32(
                    false, afrag[k], false, bfrag[k],
                    (short)0, acc, false, false);
            }
            // Epilogue: D + x -> out1 (streaming, non-temporal: never re-read)
#pragma unroll
            for (int m = 0; m < 8; ++m) {
                const int i = it * 16 + mbase + m;
                const size_t g = obase + (size_t)i * T_DIM + t0;
                __builtin_nontemporal_store(acc[m] + x[g], &out1[g]);
            }
        }
    }
}

extern "C" void kernel_launch(void* const* d_in, const int* in_sizes, int n_in,
                              void* d_out, int out_size, void* d_ws, size_t ws_size,
                              hipStream_t stream) {
    (void)in_sizes; (void)n_in; (void)out_size; (void)d_ws; (void)ws_size;
    const float* x  = (const float*)d_in[0];
    const float* wq = (const float*)d_in[1];
    const float* bq = (const float*)d_in[2];
    const float* wk = (const float*)d_in[3];
    const float* bk = (const float*)d_in[4];
    float* out1 = (float*)d_out;
    float* out2 = out1 + (size_t)256 * 64 * 4096;   // second tuple output
    dga_kernel<<<256, 256, 0, stream>>>(x, wq, bq, wk, bk, out1, out2);
}